// SINDyModel_87428354277603
// MI455X (gfx1250) — compile-verified
//
#include <hip/hip_runtime.h>
#include <hip/hip_bf16.h>

// ---------------------------------------------------------------------------
// SINDy forward:  z_dot = theta(z) @ (C * mask),  l1 = mean(|C*mask|)
//   z:     (N=262144, 16) f32
//   theta: (N, 969) = [1, z_i, z_i z_j (i<=j), z_i z_j z_k (i<=j<=k)]
//   C,mask:(969, 16) f32
//
// Compute-bound (8.1 GFLOP f32 GEMM vs 32 MB traffic @ 23.3 TB/s -> 1.4 us).
// f32 WMMA 16x16x4 keeps exact f32 precision at matrix-pipe rates.
//   - One wave = one 16-row tile; 243 chained WMMAs march K over the padded
//     972-column library.
//   - A fragments synthesized in registers from per-lane z values
//     (lane-half cndmask selects + muls; adjacent-step products CSE'd).
//   - B staged in LDS in *fragment order*: one aligned ds_load_b64 per step
//     with an immediate offset (256 B LDS traffic per WMMA, matching the
//     256 B/cycle/WGP LDS rate at the 1 WMMA/cycle/WGP matrix rate).
//   - The B base offset is laundered through an empty asm per tile iteration
//     so LICM cannot hoist the 486 fragment registers out of the loop
//     (round 1: >256 VGPRs + s_set_vgpr_msb spam; round 2: scratch spills).
//   - Low register footprint -> >=4 waves/SIMD, overlapping accumulator
//     chains keep the matrix pipe busy.
// ---------------------------------------------------------------------------

typedef __attribute__((ext_vector_type(2))) float v2f;
typedef __attribute__((ext_vector_type(8))) float v8f;

#define LATENT     16
#define LIB_DIM    969              // 1 + 16 + 136 + 816
#define LIB_PAD    972              // padded to multiple of 4 (K per WMMA step)
#define NSTEPS     (LIB_PAD / 4)    // 243 WMMA K-steps
#define COEF_ELEMS (LIB_DIM * 16)   // 15504
#define LDS_ELEMS  (LIB_PAD * 16)   // 15552

// Component index for monomial of library column `col`, component `which` (0..2).
// Returns 0..15 -> z component, 16 -> multiply by 1.0, 17 -> 0.0 (zero pad).
constexpr int monIdx(int col, int which) {
  if (col >= LIB_DIM) return 17;                 // zero padding columns
  if (col == 0) return 16;                       // constant 1
  if (col <= 16) return (which == 0) ? (col - 1) : 16;   // linear z_i
  if (col <= 152) {                              // pairs, triu order
    int t = col - 17;
    int i = 0;
    while (t >= LATENT - i) { t -= LATENT - i; ++i; }
    return (which == 0) ? i : ((which == 1) ? (i + t) : 16);
  }
  int t = col - 153;                             // triples, i<=j<=k lexicographic
  for (int i = 0; i < LATENT; ++i)
    for (int j = i; j < LATENT; ++j) {
      int run = LATENT - j;
      if (t < run) return (which == 0) ? i : ((which == 1) ? j : (j + t));
      t -= run;
    }
  return 17;
}

// Fully-unrolled K-march. Template parameter makes every monomial index a
// compile-time constant, so zr[] indexing stays in registers and lane-half
// selection lowers to v_cndmask_b32 (folds away when both halves agree).
template <int S>
struct StepRun {
  static __device__ __attribute__((always_inline)) void go(
      const float (&zr)[18], const float* bp, bool hi, v8f& acc) {
    constexpr int c = 4 * S;
    constexpr int a0l = monIdx(c + 0, 0), b0l = monIdx(c + 0, 1), c0l = monIdx(c + 0, 2);
    constexpr int a0h = monIdx(c + 2, 0), b0h = monIdx(c + 2, 1), c0h = monIdx(c + 2, 2);
    constexpr int a1l = monIdx(c + 1, 0), b1l = monIdx(c + 1, 1), c1l = monIdx(c + 1, 2);
    constexpr int a1h = monIdx(c + 3, 0), b1h = monIdx(c + 3, 1), c1h = monIdx(c + 3, 2);

    // B fragment for this step: fragment-ordered LDS layout makes the two
    // per-lane values adjacent -> single aligned ds_load_b64, imm offset S*256.
    v2f b = *(const v2f*)(bp + S * 64);

    // A fragment: theta columns for this lane's row, synthesized in-register.
    v2f a;
    a.x = (hi ? zr[a0h] : zr[a0l]) * (hi ? zr[b0h] : zr[b0l]) * (hi ? zr[c0h] : zr[c0l]);
    a.y = (hi ? zr[a1h] : zr[a1l]) * (hi ? zr[b1h] : zr[b1l]) * (hi ? zr[c1h] : zr[c1l]);

    acc = __builtin_amdgcn_wmma_f32_16x16x4_f32(
        /*neg_a=*/false, a, /*neg_b=*/false, b,
        /*c_mod=*/(short)0, acc, /*reuse_a=*/false, /*reuse_b=*/false);

    StepRun<S + 1>::go(zr, bp, hi, acc);
  }
};
template <>
struct StepRun<NSTEPS> {
  static __device__ __attribute__((always_inline)) void go(
      const float (&)[18], const float*, bool, v8f&) {}
};

__global__ __launch_bounds__(256)
__attribute__((amdgpu_waves_per_eu(4)))
void sindy_wmma_kernel(
    const float* __restrict__ z, const float* __restrict__ coef,
    const float* __restrict__ mask, float* __restrict__ out, int nRows) {
  // Masked coefficients staged in LDS in fragment order (62208 B):
  //   element (row r, col n) -> Blds[S*64 + h*32 + n*2 + p]
  //   with S = r>>2 (K-step), h = (r>>1)&1 (lane half), p = r&1 (b.x/b.y).
  __shared__ float Blds[LDS_ELEMS];
  for (int i = threadIdx.x; i < LDS_ELEMS; i += blockDim.x) {
    const int r = i >> 4, n = i & 15;
    const int S = r >> 2, h = (r >> 1) & 1, p = r & 1;
    const float v = (r < LIB_DIM) ? coef[i] * mask[i] : 0.0f;
    Blds[S * 64 + h * 32 + n * 2 + p] = v;
  }
  __syncthreads();

  const int lane = threadIdx.x & 31;
  const int n = lane & 15;                 // output column / tile row index
  const bool hi = (lane & 16) != 0;        // upper lane half (K+2 / M+8)
  const unsigned boff0 = (hi ? 32u : 0u) + (unsigned)n * 2u;

  const int wavesTotal = gridDim.x * (blockDim.x >> 5);
  const int wave = blockIdx.x * (blockDim.x >> 5) + (threadIdx.x >> 5);
  const int tilesTotal = nRows >> 4;

  // Wave-uniform grid-stride over 16-row tiles (EXEC stays all-ones: WMMA req).
  for (int tile = wave; tile < tilesTotal; tile += wavesTotal) {
    const int rowbase = tile << 4;

    // Launder the B offset so LICM cannot prove the 243 LDS loads
    // loop-invariant (prevents hoisting B into 486 VGPRs / scratch spills).
    unsigned boff = boff0;
    asm volatile("" : "+v"(boff));
    const float* bp = &Blds[boff];        // stays addrspace(3) -> ds_load_b64

    // Each lane holds the full z row it contributes to the A fragment
    // (lanes L and L+16 both own row L, per the 16x4 f32 A layout).
    float zr[18];
    const float* zrow = z + (size_t)(rowbase + n) * LATENT;
#pragma unroll
    for (int q = 0; q < LATENT; q += 4) {
      float4 t4 = *(const float4*)(zrow + q);   // global_load_b128
      zr[q] = t4.x; zr[q + 1] = t4.y; zr[q + 2] = t4.z; zr[q + 3] = t4.w;
    }
    zr[16] = 1.0f;   // identity component (x*1.0 folds away)
    zr[17] = 0.0f;   // zero-pad component

    v8f acc = {0.f, 0.f, 0.f, 0.f, 0.f, 0.f, 0.f, 0.f};
    StepRun<0>::go(zr, bp, hi, acc);

    // D layout: VGPR v, lanes 0-15 -> (M=v, N=lane); lanes 16-31 -> (M=v+8).
    float* op = out + (size_t)(rowbase + (hi ? 8 : 0)) * 16 + n;
#pragma unroll
    for (int v = 0; v < 8; ++v) op[v * 16] = acc[v];  // 8x coalesced b32 stores
  }
}

// Deterministic single-block L1 reduction: mean(|C*mask|) over 15504 elems.
__global__ __launch_bounds__(256) void sindy_l1_kernel(
    const float* __restrict__ coef, const float* __restrict__ mask,
    float* __restrict__ out_scalar) {
  __shared__ float red[256];
  float s = 0.0f;
  for (int i = threadIdx.x; i < COEF_ELEMS; i += 256)
    s += fabsf(coef[i] * mask[i]);
  red[threadIdx.x] = s;
  __syncthreads();
#pragma unroll
  for (int w = 128; w > 0; w >>= 1) {
    if ((int)threadIdx.x < w) red[threadIdx.x] += red[threadIdx.x + w];
    __syncthreads();
  }
  if (threadIdx.x == 0) out_scalar[0] = red[0] / (float)COEF_ELEMS;
}

extern "C" void kernel_launch(void* const* d_in, const int* in_sizes, int n_in,
                              void* d_out, int out_size, void* d_ws, size_t ws_size,
                              hipStream_t stream) {
  const float* z    = (const float*)d_in[0];   // (N,16) f32
  const float* coef = (const float*)d_in[1];   // (969,16) f32
  const float* mask = (const float*)d_in[2];   // (969,16) f32
  float* out = (float*)d_out;                  // N*16 f32 z_dot + 1 f32 loss

  const int nRows = in_sizes[0] / LATENT;      // 262144
  const int tiles = nRows >> 4;                // 16384

  // 8 waves/block; target ~4 tiles per wave (grid-stride handles remainder).
  int blocks = tiles / (8 * 4);
  if (blocks < 1) blocks = 1;

  sindy_wmma_kernel<<<blocks, 256, 0, stream>>>(z, coef, mask, out, nRows);
  sindy_l1_kernel<<<1, 256, 0, stream>>>(coef, mask, out + (size_t)nRows * 16);
}